// MiniGNN_80496277061654
// MI455X (gfx1250) — compile-verified
//
#include <hip/hip_runtime.h>

typedef __attribute__((ext_vector_type(2))) float v2f;
typedef __attribute__((ext_vector_type(8))) float v8f;

// ---------------------------------------------------------------------------
// Degree / normalization kernels
// ---------------------------------------------------------------------------
__global__ void init_ones_kernel(float* __restrict__ deg, int n) {
    int i = blockIdx.x * blockDim.x + threadIdx.x;
    if (i < n) deg[i] = 1.0f;  // self-loop contributes 1 to every node's degree
}

__global__ void degree_kernel(const long long* __restrict__ dst, int E,
                              float* __restrict__ deg) {
    int e = blockIdx.x * blockDim.x + threadIdx.x;
    if (e < E) atomicAdd(&deg[(int)dst[e]], 1.0f);
}

__global__ void rsqrt_kernel(float* __restrict__ d, int n) {
    int i = blockIdx.x * blockDim.x + threadIdx.x;
    if (i < n) d[i] = rsqrtf(d[i]);  // deg >= 1 always (self loop)
}

// ---------------------------------------------------------------------------
// Dense GEMM via V_WMMA_F32_16X16X4_F32 (wave32, one wave per 16-row tile).
// C[16 x NOUT] = A[16 x K] * B[K x NOUT], M must be a multiple of 16.
// Layout (ISA 7.12.2, 32-bit operands):
//   A: lane L -> row m = L%16, k-half kh = L/16; v2f = A[m][k + 2*kh + {0,1}]
//   B: lane L -> col n = L%16;                    v2f = B[k + 2*kh + {0,1}][n]
//   C/D: vgpr j on lane L -> row 8*kh + j, col n = L%16
// ---------------------------------------------------------------------------
template <int K, int NOUT>
__global__ __launch_bounds__(32) void gemm_wmma_f32_kernel(
    const float* __restrict__ A, const float* __restrict__ B,
    float* __restrict__ C) {
    constexpr int NT = NOUT / 16;
    const int rt   = blockIdx.x;        // 16-row tile index
    const int lane = threadIdx.x;       // 0..31
    const int m    = lane & 15;
    const int kh   = lane >> 4;         // 0 or 1

    const float* arow = A + (size_t)(rt * 16 + m) * K + 2 * kh;

    v8f acc[NT];
#pragma unroll
    for (int t = 0; t < NT; ++t) acc[t] = (v8f){0, 0, 0, 0, 0, 0, 0, 0};

#pragma unroll 4
    for (int k = 0; k < K; k += 4) {
        v2f a = *(const v2f*)(arow + k);  // 8B-aligned: row*K + 2*kh + k is even
#pragma unroll
        for (int t = 0; t < NT; ++t) {
            const float* bp = B + (size_t)(k + 2 * kh) * NOUT + t * 16 + m;
            v2f b;
            b.x = bp[0];
            b.y = bp[NOUT];
            acc[t] = __builtin_amdgcn_wmma_f32_16x16x4_f32(
                /*neg_a=*/false, a, /*neg_b=*/false, b,
                /*c_mod=*/(short)0, acc[t], /*reuse_a=*/false, /*reuse_b=*/false);
        }
    }

#pragma unroll
    for (int t = 0; t < NT; ++t) {
#pragma unroll
        for (int j = 0; j < 8; ++j) {
            int row = rt * 16 + kh * 8 + j;
            C[(size_t)row * NOUT + t * 16 + m] = acc[t][j];
        }
    }
}

// ---------------------------------------------------------------------------
// Self-loop init: out[i][f] = h[i][f] * dinv[i]^2 (+ bias[f] if given)
// ---------------------------------------------------------------------------
template <int F, bool WITH_BIAS>
__global__ void selfloop_init_kernel(const float* __restrict__ h,
                                     const float* __restrict__ dinv,
                                     const float* __restrict__ bias,
                                     float* __restrict__ out, int n) {
    int idx = blockIdx.x * blockDim.x + threadIdx.x;
    if (idx >= n * F) return;
    int i = idx / F;
    int f = idx - i * F;
    float di = dinv[i];
    float v  = h[idx] * di * di;
    if (WITH_BIAS) v += bias[f];
    out[idx] = v;
}

// ---------------------------------------------------------------------------
// Edge scatter: out[dst] += h[src] * dinv[src] * dinv[dst]
// One thread handles 4 features of one edge (float4 gather + 4 f32 atomics).
// ---------------------------------------------------------------------------
template <int F>
__global__ void scatter_edges_kernel(const long long* __restrict__ ei, int E,
                                     const float* __restrict__ h,
                                     const float* __restrict__ dinv,
                                     float* __restrict__ out) {
    constexpr int TPE = F / 4;  // threads per edge
    long long gid = (long long)blockIdx.x * blockDim.x + threadIdx.x;
    long long e   = gid / TPE;
    if (e >= E) return;
    int f   = (int)(gid - e * TPE) * 4;
    int src = (int)ei[e];
    int dst = (int)ei[(long long)E + e];
    float norm = dinv[src] * dinv[dst];
    const float4 v = *(const float4*)(h + (size_t)src * F + f);
    float* o = out + (size_t)dst * F + f;
    atomicAdd(o + 0, v.x * norm);
    atomicAdd(o + 1, v.y * norm);
    atomicAdd(o + 2, v.z * norm);
    atomicAdd(o + 3, v.w * norm);
}

// ---------------------------------------------------------------------------
// ReLU + bias: out[i][f] = max(a[i][f] + bias[f], 0)
// ---------------------------------------------------------------------------
template <int F>
__global__ void relu_bias_kernel(const float* __restrict__ a,
                                 const float* __restrict__ bias,
                                 float* __restrict__ out, int n) {
    int idx = blockIdx.x * blockDim.x + threadIdx.x;
    if (idx >= n * F) return;
    int f = idx % F;
    float v = a[idx] + bias[f];
    out[idx] = v > 0.0f ? v : 0.0f;
}

static inline int cdiv(long long a, long long b) { return (int)((a + b - 1) / b); }

extern "C" void kernel_launch(void* const* d_in, const int* in_sizes, int n_in,
                              void* d_out, int out_size, void* d_ws, size_t ws_size,
                              hipStream_t stream) {
    const float*     x  = (const float*)d_in[0];      // [N, 256]
    const long long* ei = (const long long*)d_in[1];  // [2, E] int64
    const float*     W1 = (const float*)d_in[2];      // [256, 64]
    const float*     b1 = (const float*)d_in[3];      // [64]
    const float*     W2 = (const float*)d_in[4];      // [64, 128]
    const float*     b2 = (const float*)d_in[5];      // [128]
    float*           out = (float*)d_out;             // [N, 128]

    const int IN = 256, HID = 64, OUT = 128;
    const int N = in_sizes[0] / IN;      // 100000 (multiple of 16)
    const int E = in_sizes[1] / 2;       // 1600000

    float* ws   = (float*)d_ws;
    float* dinv = ws;                                  // [N]
    float* h1   = dinv + N;                            // [N, 64]
    float* a1   = h1 + (size_t)N * HID;                // [N, 64]
    float* h3   = a1 + (size_t)N * HID;                // [N, 128]

    const int B256 = 256;

    // 1) degrees (dst side, + self loop) -> dinv = rsqrt(deg)
    init_ones_kernel<<<cdiv(N, B256), B256, 0, stream>>>(dinv, N);
    degree_kernel<<<cdiv(E, B256), B256, 0, stream>>>(ei + E, E, dinv);
    rsqrt_kernel<<<cdiv(N, B256), B256, 0, stream>>>(dinv, N);

    // 2) h1 = x @ W1   (WMMA f32, one wave per 16 rows)
    gemm_wmma_f32_kernel<IN, HID><<<N / 16, 32, 0, stream>>>(x, W1, h1);

    // 3) a1 = self-loop term, then scatter edges, then ReLU+bias -> h1 (reused)
    selfloop_init_kernel<HID, false>
        <<<cdiv((long long)N * HID, B256), B256, 0, stream>>>(h1, dinv, nullptr, a1, N);
    scatter_edges_kernel<HID>
        <<<cdiv((long long)E * (HID / 4), B256), B256, 0, stream>>>(ei, E, h1, dinv, a1);
    relu_bias_kernel<HID>
        <<<cdiv((long long)N * HID, B256), B256, 0, stream>>>(a1, b1, h1, N);

    // 4) h3 = relu_out @ W2
    gemm_wmma_f32_kernel<HID, OUT><<<N / 16, 32, 0, stream>>>(h1, W2, h3);

    // 5) out = self-loop term + b2, then scatter edges into out
    selfloop_init_kernel<OUT, true>
        <<<cdiv((long long)N * OUT, B256), B256, 0, stream>>>(h3, dinv, b2, out, N);
    scatter_edges_kernel<OUT>
        <<<cdiv((long long)E * (OUT / 4), B256), B256, 0, stream>>>(ei, E, h3, dinv, out);
}